// CMoERouter_51427938402768
// MI455X (gfx1250) — compile-verified
//
#include <hip/hip_runtime.h>

typedef __attribute__((ext_vector_type(2))) float v2f;
typedef __attribute__((ext_vector_type(4))) float v4f;
typedef __attribute__((ext_vector_type(8))) float v8f;

#define D_DIM 1024
#define K_CENT 8
#define ROWS_PER_WAVE 16
#define WAVES_PER_BLOCK 8
#define ROWS_PER_BLOCK (ROWS_PER_WAVE * WAVES_PER_BLOCK)

__global__ __launch_bounds__(256)
void cmoe_router_kernel(const float* __restrict__ x,
                        const float* __restrict__ centroids,
                        float* __restrict__ weights,
                        int* __restrict__ assignments) {
    // Stage centroids (8 x 1024 f32 = 32KB) in LDS, coalesced 16B copies.
    __shared__ float lds_c[K_CENT * D_DIM];
    {
        const v4f* src = (const v4f*)centroids;
        v4f*       dst = (v4f*)lds_c;
        const int nvec = (K_CENT * D_DIM) / 4; // 2048
        for (int i = threadIdx.x; i < nvec; i += blockDim.x) dst[i] = src[i];
    }
    __syncthreads();

    const int lane = threadIdx.x & 31;
    const int wave = threadIdx.x >> 5;
    const int half = lane >> 4;   // selects K-pair (0: k,k+1  1: k+2,k+3)
    const int nl   = lane & 15;   // A: local row  /  B,C,D: column index
    const int row0 = blockIdx.x * ROWS_PER_BLOCK + wave * ROWS_PER_WAVE;
    const int myrow = row0 + nl;

    // A fragment source: x[myrow][k + 2*half .. +1]   (8B aligned)
    const float* xbase = x + (size_t)myrow * D_DIM + 2 * half;
    // B fragment source: centroids[nl][k + 2*half .. +1]; columns 8..15 -> 0
    const float* cbase = lds_c + (size_t)(nl & (K_CENT - 1)) * D_DIM + 2 * half;
    const float  bmask = (nl < K_CENT) ? 1.0f : 0.0f;

    v8f   acc = {};          // 16x16 f32 accumulator: acc[j] = dot(row j+8*half, cent nl)
    float xm2 = 0.0f;        // partial ||x_row||^2 (this lane's K-offsets)
    float cn2 = 0.0f;        // partial ||c_col||^2

#pragma unroll 8
    for (int k = 0; k < D_DIM; k += 4) {
        v2f a = *(const v2f*)(xbase + k);   // global_load_b64, WMMA A layout
        v2f b = *(const v2f*)(cbase + k);   // ds_load_b64,    WMMA B layout
        b.x *= bmask;
        b.y *= bmask;
        xm2 = __builtin_fmaf(a.x, a.x, xm2);
        xm2 = __builtin_fmaf(a.y, a.y, xm2);
        cn2 = __builtin_fmaf(b.x, b.x, cn2);
        cn2 = __builtin_fmaf(b.y, b.y, cn2);
        // D = A(16x4 f32) * B(4x16 f32) + C  -- exact fp32 on the matrix pipe
        acc = __builtin_amdgcn_wmma_f32_16x16x4_f32(
            /*neg_a=*/false, a, /*neg_b=*/false, b,
            /*c_mod=*/(short)0, acc, /*reuse_a=*/false, /*reuse_b=*/false);
    }
    // Combine the two K-halves: lane L pairs with lane L^16.
    xm2 += __shfl_xor(xm2, 16);
    cn2 += __shfl_xor(cn2, 16);

#pragma unroll
    for (int j = 0; j < 8; ++j) {
        const int   m    = j + 8 * half;           // local row for this C VGPR
        const float xr   = __shfl(xm2, m);         // broadcast row norm
        const float sq   = xr + cn2 - 2.0f * acc[j];
        const float dist = __builtin_sqrtf(fmaxf(sq, 0.0f));

        // min + first-index argmin over n = 0..7 (butterfly stays in groups of 8)
        float dmin = dist;
        int   imin = nl;
#pragma unroll
        for (int mask = 1; mask <= 4; mask <<= 1) {
            const float od = __shfl_xor(dmin, mask);
            const int   oi = __shfl_xor(imin, mask);
            if (od < dmin || (od == dmin && oi < imin)) { dmin = od; imin = oi; }
        }

        // softmax(-dist) = exp(dmin - dist) / sum
        const float e  = __expf(dmin - dist);
        float       es = e;
#pragma unroll
        for (int mask = 1; mask <= 4; mask <<= 1) es += __shfl_xor(es, mask);
        const float w = e / es;

        const int grow = row0 + m;
        if (nl < K_CENT) weights[(size_t)grow * K_CENT + nl] = w;
        if (nl == 0)     assignments[grow] = imin;
    }
}

extern "C" void kernel_launch(void* const* d_in, const int* in_sizes, int n_in,
                              void* d_out, int out_size, void* d_ws, size_t ws_size,
                              hipStream_t stream) {
    const float* x = (const float*)d_in[0];        // (16, 4096, 1024) f32
    const float* c = (const float*)d_in[1];        // (8, 1024) f32
    const int nrows = in_sizes[0] / D_DIM;         // 65536

    float* weights = (float*)d_out;                            // (nrows, 8) f32
    int*   assign  = (int*)d_out + (size_t)nrows * K_CENT;     // (nrows,) i32 bits

    const int blocks = nrows / ROWS_PER_BLOCK;     // 512
    hipLaunchKernelGGL(cmoe_router_kernel, dim3(blocks), dim3(256), 0, stream,
                       x, c, weights, assign);
}